// MambaLayer_3dlocalcube_36326833389653
// MI455X (gfx1250) — compile-verified
//
#include <hip/hip_runtime.h>
#include <math.h>

typedef __attribute__((ext_vector_type(2))) float v2f;
typedef __attribute__((ext_vector_type(8))) float v8f;

#define BLK     512
#define NWAVES  16
#define TCH     128          // tokens per chunk
#define LTOT    1025         // 1 gating token + 1024

// ---- LDS layout (float offsets) ----
#define O_WIN   0            // in_w        64x16     = 1024
#define O_WXP   1024         // xproj padded 48x32    = 1536
#define O_WOUT  2560         // out_w       16x32     = 512
#define O_CONVW 3072         // 32x4 = 128
#define O_CONVB 3200         // 32
#define O_DTW   3232         // 32
#define O_DTB   3264         // 32
#define O_NEGA  3296         // -exp(A_log) 32x16 = 512
#define O_DP    3808         // 32
#define O_LNG   3840         // 16
#define O_LNB   3856         // 16
#define O_GT    3872         // 16
#define O_XLN   3888         // TCH x 16  = 2048
#define O_XI    5936         // (TCH+3) x 32 = 4192  (3-row causal conv halo at front)
#define O_Z     10128        // TCH x 32 = 4096
#define O_XC    14224        // TCH x 32 = 4096
#define O_XDBL  18320        // TCH x 48 = 6144 (col0=dt, 1..16=B, 17..32=C)
#define O_DTF   24464        // TCH x 32 = 4096 (softplus dt, later reused as gated y)
#define O_YV    28560        // TCH x 32 = 4096
#define O_OUT   32656        // 64x16x16 = 16384
#define SMEM_FLOATS 49040    // = 196160 bytes

struct MWeights {
  const float *in_w, *conv_w, *conv_b, *xproj_w, *dt_w, *dt_b, *A_log, *Dp, *out_w;
};

__device__ __forceinline__ float lane_sum16(float v) {
  v += __shfl_xor(v, 1);
  v += __shfl_xor(v, 2);
  v += __shfl_xor(v, 4);
  v += __shfl_xor(v, 8);
  return v;
}

// one 16x16 C tile, K steps of V_WMMA_F32_16X16X4_F32
// A: row-major [16 x K] at lds stride 'lda'; Bsrc: weight rows [n][k] at stride 'ldb'
__device__ __forceinline__ v8f wmma_tile(const float* __restrict__ A, int lda,
                                         const float* __restrict__ Bw, int ldb,
                                         int K, int mIdx, int half) {
  v8f c = {};
  #pragma unroll
  for (int k0 = 0; k0 < 32; k0 += 4) {
    if (k0 >= K) break;
    v2f a, bb;
    a.x  = A[mIdx * lda + k0 + 2 * half];
    a.y  = A[mIdx * lda + k0 + 1 + 2 * half];
    bb.x = Bw[mIdx * ldb + k0 + 2 * half];
    bb.y = Bw[mIdx * ldb + k0 + 1 + 2 * half];
    c = __builtin_amdgcn_wmma_f32_16x16x4_f32(false, a, false, bb, (short)0, c, false, false);
  }
  return c;
}

__global__ __launch_bounds__(BLK)
void mamba_cube_kernel(const float* __restrict__ X,
                       const float* __restrict__ gt1, const float* __restrict__ gt2,
                       const float* __restrict__ ln1g, const float* __restrict__ ln1b,
                       const float* __restrict__ ln2g, const float* __restrict__ ln2b,
                       MWeights m1, MWeights m2,
                       float* __restrict__ Out)
{
  extern __shared__ float sm[];
  const int tid  = threadIdx.x;
  const int lane = tid & 31;
  const int wid  = tid >> 5;
  const int mIdx = lane & 15;       // M (or N) index inside WMMA tile
  const int half = lane >> 4;       // lane half select

  const int n    = blockIdx.x;      // sequence id 0..127
  const int b    = n >> 6;
  const int cube = n & 63;
  const int ci   = cube >> 3, cj = cube & 7;

  // zero output accumulator (shared by all 3 modes)
  for (int i = tid; i < 16384; i += BLK) sm[O_OUT + i] = 0.f;

  for (int mode = 0; mode < 3; ++mode) {
    const MWeights W   = (mode == 0) ? m1 : m2;
    const float*   lng = (mode == 0) ? ln1g : ln2g;
    const float*   lnb = (mode == 0) ? ln1b : ln2b;
    const float*   gtp = (mode == 0) ? gt1 : gt2;

    __syncthreads();
    // ---- stage weights for this mode ----
    for (int i = tid; i < 1024; i += BLK) sm[O_WIN + i] = W.in_w[i];
    for (int i = tid; i < 1536; i += BLK) {
      int nn = i >> 5;                      // xproj stored [n][k], rows 33..47 zero pad
      sm[O_WXP + i] = (nn < 33) ? W.xproj_w[i] : 0.f;
    }
    for (int i = tid; i < 512; i += BLK) sm[O_WOUT + i] = W.out_w[i];
    for (int i = tid; i < 512; i += BLK) sm[O_NEGA + i] = -__expf(W.A_log[i]);
    if (tid < 128) sm[O_CONVW + tid] = W.conv_w[tid];
    if (tid < 32) {
      sm[O_CONVB + tid] = W.conv_b[tid];
      sm[O_DTW  + tid]  = W.dt_w[tid];
      sm[O_DTB  + tid]  = W.dt_b[tid];
      sm[O_DP   + tid]  = W.Dp[tid];
    }
    if (tid < 16) {
      sm[O_LNG + tid] = lng[tid];
      sm[O_LNB + tid] = lnb[tid];
      sm[O_GT  + tid] = gtp[tid];
    }
    if (tid < 96) sm[O_XI + tid] = 0.f;     // zero causal-conv halo
    __syncthreads();

    const float negA = sm[O_NEGA + tid];    // tid == d*16+s exactly
    const float dval = sm[O_DP + (tid >> 4)];
    float h = 0.f;                          // scan state for (d = tid>>4, s = tid&15)

    for (int t0 = 0; t0 < LTOT; t0 += TCH) {
      const int ltmax = (LTOT - t0 < TCH) ? (LTOT - t0) : TCH;

      // ---- gather + LayerNorm: 16 lanes per token ----
      for (int p = 0; p < TCH / 32; ++p) {
        int tok = p * 32 + wid * 2 + half;  // 32 tokens per pass
        int f   = mIdx;
        int t   = t0 + tok;
        bool valid = (t < LTOT);
        float v = 0.f;
        if (valid) {
          if (t == 0) {
            v = sm[O_GT + f];
          } else {
            int l = t - 1, c, hh, ww;
            if (mode == 0) {                // pixelshuffle(r=2) view, d = c/4 group
              int hr = l >> 5, wr = l & 31;
              c = f * 4 + (hr & 1) * 2 + (wr & 1); hh = hr >> 1; ww = wr >> 1;
            } else if (mode == 1) {         // d = spatial row
              c = l >> 4; ww = l & 15; hh = f;
            } else {                        // d = spatial col
              c = l >> 4; hh = l & 15; ww = f;
            }
            v = X[(size_t)((b * 64 + c) * 128 + ci * 16 + hh) * 128 + cj * 16 + ww];
          }
        }
        float mu = lane_sum16(v) * (1.f / 16.f);
        float dv = v - mu;
        float var = lane_sum16(dv * dv) * (1.f / 16.f);
        float res = valid ? (dv * rsqrtf(var + 1e-5f) * sm[O_LNG + f] + sm[O_LNB + f]) : 0.f;
        sm[O_XLN + tok * 16 + f] = res;
      }
      __syncthreads();

      // ---- GEMM1: xz = xln(Tx16) @ in_w^T(16x64) via WMMA f32 16x16x4 ----
      for (int job = wid; job < (TCH / 16) * 4; job += NWAVES) {
        int mt = job >> 2, nt = job & 3;
        v8f c = wmma_tile(&sm[O_XLN + (mt * 16) * 16], 16,
                          &sm[O_WIN + (nt * 16) * 16], 16, 16, mIdx, half);
        // nt is wave-uniform: nt<2 -> xi columns, else z columns (no lane divergence)
        if (nt < 2) {
          float* dst = &sm[O_XI + (3 + mt * 16 + 8 * half) * 32 + nt * 16 + mIdx];
          #pragma unroll
          for (int r = 0; r < 8; ++r) dst[r * 32] = c[r];
        } else {
          float* dst = &sm[O_Z + (mt * 16 + 8 * half) * 32 + (nt - 2) * 16 + mIdx];
          #pragma unroll
          for (int r = 0; r < 8; ++r) dst[r * 32] = c[r];
        }
      }
      __syncthreads();

      // ---- causal depthwise conv(4) + SiLU ----
      for (int i = tid; i < TCH * 32; i += BLK) {
        int lt = i >> 5, d = i & 31;
        float acc = sm[O_CONVB + d];
        #pragma unroll
        for (int j = 0; j < 4; ++j)
          acc += sm[O_CONVW + d * 4 + j] * sm[O_XI + (lt + j) * 32 + d];
        sm[O_XC + i] = acc / (1.f + __expf(-acc));
      }
      __syncthreads();
      // carry conv halo (last 3 raw xi rows) into front slots for next chunk
      if (tid < 96) sm[O_XI + tid] = sm[O_XI + TCH * 32 + tid];

      // ---- GEMM2: x_dbl = xc(Tx32) @ xproj^T(32x33, padded to 48) ----
      for (int job = wid; job < (TCH / 16) * 3; job += NWAVES) {
        int mt = job / 3, nt = job % 3;
        v8f c = wmma_tile(&sm[O_XC + (mt * 16) * 32], 32,
                          &sm[O_WXP + (nt * 16) * 32], 32, 32, mIdx, half);
        float* dst = &sm[O_XDBL + (mt * 16 + 8 * half) * 48 + nt * 16 + mIdx];
        #pragma unroll
        for (int r = 0; r < 8; ++r) dst[r * 48] = c[r];
      }
      __syncthreads();

      // ---- dt = softplus(x_dbl[:,0] * dt_w + dt_b) ----
      for (int i = tid; i < TCH * 32; i += BLK) {
        int lt = i >> 5, d = i & 31;
        float raw = sm[O_XDBL + lt * 48] * sm[O_DTW + d] + sm[O_DTB + d];
        sm[O_DTF + i] = (raw > 20.f) ? raw : log1pf(__expf(raw));
      }
      __syncthreads();

      // ---- selective scan: thread = (d,s); sequential over chunk tokens ----
      {
        const int d = tid >> 4, s = tid & 15;
        for (int lt = 0; lt < ltmax; ++lt) {
          float dtv = sm[O_DTF  + lt * 32 + d];
          float xcv = sm[O_XC   + lt * 32 + d];
          float bm  = sm[O_XDBL + lt * 48 + 1 + s];
          float cm  = sm[O_XDBL + lt * 48 + 17 + s];
          float a   = __expf(dtv * negA);
          h = a * h + dtv * bm * xcv;
          float p = h * cm;
          p += __shfl_xor(p, 1);
          p += __shfl_xor(p, 2);
          p += __shfl_xor(p, 4);
          p += __shfl_xor(p, 8);
          if (s == 0) sm[O_YV + lt * 32 + d] = p + dval * xcv;
        }
      }
      __syncthreads();

      // ---- gate: y *= silu(z)  (reuse DTF as gated-y buffer) ----
      for (int i = tid; i < TCH * 32; i += BLK) {
        float zv = sm[O_Z + i];
        sm[O_DTF + i] = sm[O_YV + i] * (zv / (1.f + __expf(-zv)));
      }
      __syncthreads();

      // ---- GEMM3: out = y(Tx32) @ out_w^T(32x16), scatter into cube ----
      for (int job = wid; job < TCH / 16; job += NWAVES) {
        int mt = job;
        v8f c = wmma_tile(&sm[O_DTF + (mt * 16) * 32], 32,
                          &sm[O_WOUT], 32, 32, mIdx, half);
        const int f = mIdx;
        // mode is uniform: three straight-line scatter variants; only the
        // t-range check (half-wave uniform) remains potentially divergent
        if (mode == 0) {
          #pragma unroll
          for (int r = 0; r < 8; ++r) {
            int t = t0 + mt * 16 + r + 8 * half;
            if (t >= 1 && t < LTOT) {
              int l = t - 1, hr = l >> 5, wr = l & 31;
              int cc = f * 4 + (hr & 1) * 2 + (wr & 1);
              sm[O_OUT + (cc * 16 + (hr >> 1)) * 16 + (wr >> 1)] += c[r];
            }
          }
        } else if (mode == 1) {
          #pragma unroll
          for (int r = 0; r < 8; ++r) {
            int t = t0 + mt * 16 + r + 8 * half;
            if (t >= 1 && t < LTOT) {
              int l = t - 1;
              sm[O_OUT + ((l >> 4) * 16 + f) * 16 + (l & 15)] += c[r];
            }
          }
        } else {
          #pragma unroll
          for (int r = 0; r < 8; ++r) {
            int t = t0 + mt * 16 + r + 8 * half;
            if (t >= 1 && t < LTOT) {
              int l = t - 1;
              sm[O_OUT + ((l >> 4) * 16 + (l & 15)) * 16 + f] += c[r];
            }
          }
        }
      }
      __syncthreads();
    } // chunks
  } // modes

  __syncthreads();
  for (int i = tid; i < 16384; i += BLK) {
    int c = i >> 8, hh = (i >> 4) & 15, ww = i & 15;
    Out[(size_t)((b * 64 + c) * 128 + ci * 16 + hh) * 128 + cj * 16 + ww] =
        sm[O_OUT + i] * (1.f / 3.f);
  }
}

extern "C" void kernel_launch(void* const* d_in, const int* in_sizes, int n_in,
                              void* d_out, int out_size, void* d_ws, size_t ws_size,
                              hipStream_t stream) {
  (void)in_sizes; (void)n_in; (void)d_ws; (void)ws_size; (void)out_size;
  const float* X    = (const float*)d_in[0];
  const float* gt1  = (const float*)d_in[1];
  const float* gt2  = (const float*)d_in[2];
  const float* ln1g = (const float*)d_in[3];
  const float* ln1b = (const float*)d_in[4];
  const float* ln2g = (const float*)d_in[5];
  const float* ln2b = (const float*)d_in[6];
  MWeights m1 = { (const float*)d_in[7],  (const float*)d_in[8],  (const float*)d_in[9],
                  (const float*)d_in[10], (const float*)d_in[11], (const float*)d_in[12],
                  (const float*)d_in[13], (const float*)d_in[14], (const float*)d_in[15] };
  MWeights m2 = { (const float*)d_in[16], (const float*)d_in[17], (const float*)d_in[18],
                  (const float*)d_in[19], (const float*)d_in[20], (const float*)d_in[21],
                  (const float*)d_in[22], (const float*)d_in[23], (const float*)d_in[24] };

  size_t shmem = (size_t)SMEM_FLOATS * sizeof(float);   // ~196 KB (WGP LDS = 320 KB)
  (void)hipFuncSetAttribute((const void*)mamba_cube_kernel,
                            hipFuncAttributeMaxDynamicSharedMemorySize, (int)shmem);
  mamba_cube_kernel<<<dim3(128), dim3(BLK), shmem, stream>>>(
      X, gt1, gt2, ln1g, ln1b, ln2g, ln2b, m1, m2, (float*)d_out);
}